// BiosyntheticCoherenceLoss_70497593196919
// MI455X (gfx1250) — compile-verified
//
#include <hip/hip_runtime.h>
#include <math.h>

// ---------------------------------------------------------------------------
// CDNA5 / gfx1250, wave32. V_WMMA_F32_16X16X4_F32 Gram matrix, symmetric
// (upper-triangle) tiling, register-resident B fragments, raw v_sqrt_f32
// epilogue, analytic same-pair count.
// ---------------------------------------------------------------------------

typedef float v2f __attribute__((ext_vector_type(2)));
typedef float v8f __attribute__((ext_vector_type(8)));

#define N_ROWS 8192
#define DIM    16

// Family id per codon index (dict insertion order of CODON_TABLE):
// glutamate=0 aspartate=1 serine=2 pyruvate=3 aromatic=4 histidine=5 stop=-1
// (K maps to aspartate: later dict-comprehension entry overwrites glutamate.)
__device__ __constant__ int FAM[64] = {
    4, 4, 3, 3, 3, 3, 3, 3,   // UUU..CUG : F F L L L L L L
    1, 1, 1, 1,               // AUU..AUG : I I I M
    3, 3, 3, 3,               // GUU..GUG : V V V V
    2, 2, 2, 2,               // UCU..UCG : S S S S
    0, 0, 0, 0,               // CCU..CCG : P P P P
    1, 1, 1, 1,               // ACU..ACG : T T T T
    3, 3, 3, 3,               // GCU..GCG : A A A A
    4, 4, -1, -1,             // UAU..UAG : Y Y * *
    5, 5, 0, 0,               // CAU..CAG : H H Q Q
    1, 1, 1, 1,               // AAU..AAG : N N K K
    1, 1, 0, 0,               // GAU..GAG : D D E E
    2, 2, -1, 4,              // UGU..UGG : C C * W
    0, 0, 0, 0,               // CGU..CGG : R R R R
    2, 2, 0, 0,               // AGU..AGG : S S R R
    2, 2, 2, 2                // GGU..GGG : G G G G
};

// --------------------------- phase 1: row prep -----------------------------
__global__ void bcl_prep(const float* __restrict__ E, const int* __restrict__ idx,
                         float* __restrict__ sq, int* __restrict__ fam) {
    int i = blockIdx.x * blockDim.x + threadIdx.x;
    if (i >= N_ROWS) return;
    const float4* p = reinterpret_cast<const float4*>(E + i * DIM);
    float s = 0.f;
#pragma unroll
    for (int j = 0; j < 4; ++j) {
        float4 v = p[j];
        s += v.x * v.x + v.y * v.y + v.z * v.z + v.w * v.w;
    }
    sq[i]  = s;
    fam[i] = FAM[idx[i] & 63];
}

// --------------------- phase 2: pairwise tiles via WMMA --------------------
// Grid: (64, 64) over 128x128 blocks of the symmetric pair matrix.
// Blocks strictly below the diagonal (bx < by) write zero partials and exit;
// blocks strictly above contribute with weight 2 (they stand for their
// transpose too); diagonal blocks contribute with weight 1.
// 8 waves/block as 2(row)x4(col); each wave owns 64x32 = 4x2 WMMA tiles.
__global__ __launch_bounds__(256)
void bcl_pairs(const float* __restrict__ E, const float* __restrict__ sq,
               const int* __restrict__ fam, double* __restrict__ partials) {
    const int bx  = blockIdx.x;
    const int by  = blockIdx.y;
    const int tid = threadIdx.x;
    const int bid = by * gridDim.x + bx;

    if (bx < by) {                     // lower triangle: no work
        if (tid == 0) { partials[bid * 2 + 0] = 0.0; partials[bid * 2 + 1] = 0.0; }
        return;
    }

    const int wave = tid >> 5;
    const int lane = tid & 31;
    const int hlf  = lane >> 4;   // 0: lanes 0-15, 1: lanes 16-31
    const int l16  = lane & 15;

    const int rb = by * 128 + (wave >> 2) * 64;   // wave row base (64 rows)
    const int cb = bx * 128 + (wave & 3) * 32;    // wave col base (32 cols)

    // Preload B fragments for the wave's 2 col blocks (16x4 f32 each).
    v2f   bf[2][4];
    float sqc[2];
    int   fmc[2];
#pragma unroll
    for (int c4 = 0; c4 < 2; ++c4) {
        const int colBase = cb + c4 * 16;
#pragma unroll
        for (int kc = 0; kc < 4; ++kc) {
            const int k0 = kc * 4 + hlf * 2;
            bf[c4][kc] = *reinterpret_cast<const v2f*>(E + (colBase + l16) * DIM + k0);
        }
        sqc[c4] = sq[colBase + l16];
        fmc[c4] = fam[colBase + l16];
    }

    float sAll = 0.f, sSame = 0.f;

    for (int r4 = 0; r4 < 4; ++r4) {
        const int rowBase = rb + r4 * 16;

        v2f a[4];
#pragma unroll
        for (int kc = 0; kc < 4; ++kc) {
            const int k0 = kc * 4 + hlf * 2;
            a[kc] = *reinterpret_cast<const v2f*>(E + (rowBase + l16) * DIM + k0);
        }
        // per-lane row-side scalars for the epilogue (C/D layout: M = r + 8*hlf)
        float sqr[8];
        int   fmr[8];
#pragma unroll
        for (int r = 0; r < 8; ++r) {
            const int m = rowBase + r + hlf * 8;
            sqr[r] = sq[m];
            fmr[r] = fam[m];
        }

#pragma unroll
        for (int c4 = 0; c4 < 2; ++c4) {
            v8f acc = {0.f, 0.f, 0.f, 0.f, 0.f, 0.f, 0.f, 0.f};
#pragma unroll
            for (int kc = 0; kc < 4; ++kc) {
                acc = __builtin_amdgcn_wmma_f32_16x16x4_f32(
                    /*neg_a=*/false, a[kc], /*neg_b=*/false, bf[c4][kc],
                    /*c_mod=*/(short)0, acc, /*reuse_a=*/false, /*reuse_b=*/false);
            }
#pragma unroll
            for (int r = 0; r < 8; ++r) {
                float d2   = sqr[r] + sqc[c4] - 2.0f * acc[r];
                float dist = __builtin_amdgcn_sqrtf(fmaxf(d2, 0.f)); // raw v_sqrt_f32
                sAll += dist;
                const bool same = (fmr[r] >= 0) && (fmr[r] == fmc[c4]);
                sSame += same ? dist : 0.f;
            }
        }
    }

    // wave32 reduction
    double dA = (double)sAll, dS = (double)sSame;
#pragma unroll
    for (int off = 16; off > 0; off >>= 1) {
        dA += __shfl_down(dA, off, 32);
        dS += __shfl_down(dS, off, 32);
    }

    __shared__ double red[8][2];
    if (lane == 0) { red[wave][0] = dA; red[wave][1] = dS; }
    __syncthreads();
    if (tid == 0) {
        double A = 0.0, S = 0.0;
#pragma unroll
        for (int w = 0; w < 8; ++w) { A += red[w][0]; S += red[w][1]; }
        const double wgt = (bx == by) ? 1.0 : 2.0;   // off-diagonal counts twice
        partials[bid * 2 + 0] = A * wgt;
        partials[bid * 2 + 1] = S * wgt;
    }
}

// ------------------------- phase 3: deterministic final --------------------
// Reduces the 4096 per-block partials and computes same_sum analytically:
// same_sum = sum_f n_f^2 over the 6-family histogram of the 8192 indices.
__global__ void bcl_finalize(const double* __restrict__ partials, int np,
                             const int* __restrict__ fam, float* __restrict__ out) {
    __shared__ double red[256][2];
    __shared__ int    hist[6];
    const int t = threadIdx.x;
    if (t < 6) hist[t] = 0;
    __syncthreads();

    for (int i = t; i < N_ROWS; i += 256) {
        const int f = fam[i];
        if (f >= 0) atomicAdd(&hist[f], 1);   // integer: deterministic
    }

    double A = 0.0, S = 0.0;
    for (int i = t; i < np; i += 256) {
        A += partials[i * 2 + 0];
        S += partials[i * 2 + 1];
    }
    red[t][0] = A; red[t][1] = S;
    __syncthreads();
    for (int s2 = 128; s2 > 0; s2 >>= 1) {
        if (t < s2) {
            red[t][0] += red[t + s2][0];
            red[t][1] += red[t + s2][1];
        }
        __syncthreads();
    }
    if (t == 0) {
        double sameCnt = 0.0;
#pragma unroll
        for (int f = 0; f < 6; ++f) {
            const double n = (double)hist[f];
            sameCnt += n * n;
        }
        const double sumAll  = red[0][0];
        const double sumSame = red[0][1];
        const double total   = (double)N_ROWS * (double)N_ROWS;
        const double eps     = 1e-10;
        const double same_d  = sumSame / (sameCnt + eps);
        const double diff_d  = (sumAll - sumSame) / (total - sameCnt + eps);
        double loss = same_d - 0.5 * diff_d + 1.0;
        out[0] = (float)(loss > 0.0 ? loss : 0.0);
    }
}

// ------------------------------- launch ------------------------------------
extern "C" void kernel_launch(void* const* d_in, const int* in_sizes, int n_in,
                              void* d_out, int out_size, void* d_ws, size_t ws_size,
                              hipStream_t stream) {
    (void)in_sizes; (void)n_in; (void)out_size; (void)ws_size;
    const float* E   = (const float*)d_in[0];   // (16,512,16) f32 -> (8192,16)
    const int*   idx = (const int*)d_in[1];     // (16,512) i32  -> (8192,)

    char*   ws       = (char*)d_ws;
    float*  sq       = (float*)(ws);            // 8192 f32  (32 KB)
    int*    fam      = (int*)(ws + 32768);      // 8192 i32  (32 KB)
    double* partials = (double*)(ws + 65536);   // 4096*2 f64 (64 KB)

    bcl_prep<<<N_ROWS / 256, 256, 0, stream>>>(E, idx, sq, fam);

    dim3 grid(64, 64);   // 128x128 pair blocks; lower triangle early-exits
    bcl_pairs<<<grid, 256, 0, stream>>>(E, sq, fam, partials);

    bcl_finalize<<<1, 256, 0, stream>>>(partials, 4096, fam, (float*)d_out);
}